// BACK_DKVMN_84146999263356
// MI455X (gfx1250) — compile-verified
//
#include <hip/hip_runtime.h>
#include <hip/hip_bf16.h>

typedef __attribute__((ext_vector_type(2))) float        v2f;
typedef __attribute__((ext_vector_type(8))) float        v8f;
typedef __attribute__((ext_vector_type(4))) unsigned int v4u;
typedef __attribute__((ext_vector_type(8))) int          v8i;
typedef __attribute__((ext_vector_type(4))) int          v4i;

#define DIMS 200
#define SLOTS 50
#define NB 64
#define NUMQ 10000

// ---------------------------------------------------------------------------
// Weight transpose + k-pair interleave: W [N,K] row-major ->
// Wt[((k>>1)*N + n)*2 + (k&1)]  => fragment (k,k+1) per column is one b64 load.
// ---------------------------------------------------------------------------
__global__ void transpose_w(const float* __restrict__ W, float* __restrict__ Wt,
                            int N, int K) {
  long i = (long)blockIdx.x * blockDim.x + threadIdx.x;
  if (i >= (long)N * K) return;
  int k = (int)(i % K);
  int n = (int)(i / K);
  Wt[((long)(k >> 1) * N + n) * 2 + (k & 1)] = W[i];
}

// ---------------------------------------------------------------------------
// Templated k-loop: AC1 = A contiguous in k (b64 fragment loads),
// MASKED = edge tile (apply zero-masks); interior waves skip all mask VALU.
// ---------------------------------------------------------------------------
template <bool AC1, bool MASKED>
__device__ __forceinline__ void gemm_kloop(
    const float* __restrict__ Ab, long aRow, int Acs,
    const v2f* __restrict__ Bp, int N, int K, int half,
    float mA, const long* cc, const float* mB,
    v8f& acc0, v8f& acc1, v8f& acc2, v8f& acc3) {
  for (int k0 = 0; k0 < K; k0 += 4) {            // K always multiple of 4 here
    int kh = (k0 >> 1) + half;                   // k-pair index: ka = 2*kh
    v2f av;
    if (AC1) {
      av = *(const v2f*)(Ab + aRow + 2L * kh);   // even offset -> 8B aligned b64
    } else {
      int ka = kh << 1;
      av.x = Ab[aRow + (long)ka * Acs];
      av.y = Ab[aRow + (long)(ka + 1) * Acs];
    }
    if (MASKED) { av.x *= mA; av.y *= mA; }
    long brow = (long)kh * N;
    v2f b0 = Bp[brow + cc[0]];                   // one b64 per fragment
    v2f b1 = Bp[brow + cc[1]];
    v2f b2 = Bp[brow + cc[2]];
    v2f b3 = Bp[brow + cc[3]];
    if (MASKED) {
      b0.x *= mB[0]; b0.y *= mB[0];
      b1.x *= mB[1]; b1.y *= mB[1];
      b2.x *= mB[2]; b2.y *= mB[2];
      b3.x *= mB[3]; b3.y *= mB[3];
    }
    acc0 = __builtin_amdgcn_wmma_f32_16x16x4_f32(false, av, false, b0, (short)0, acc0, false, false);
    acc1 = __builtin_amdgcn_wmma_f32_16x16x4_f32(false, av, false, b1, (short)0, acc1, false, false);
    acc2 = __builtin_amdgcn_wmma_f32_16x16x4_f32(false, av, false, b2, (short)0, acc2, false, false);
    acc3 = __builtin_amdgcn_wmma_f32_16x16x4_f32(false, av, false, b3, (short)0, acc3, false, false);
  }
}

// ---------------------------------------------------------------------------
// Wave-tile GEMM: C[b] = act(A[b] (MxK) @ B (KxN) + bias). B is transposed +
// k-pair packed. Each wave computes 16Mx64N: one shared A fragment feeds 4
// INDEPENDENT V_WMMA_F32_16X16X4_F32 accumulators per k-step.
// A element (r,k) at A + b*bsA + r*Ars + k*Acs (stride-parameterized transpose).
// ---------------------------------------------------------------------------
__global__ void gemm_wmma_f32(const float* __restrict__ A, long bsA, int Ars, int Acs,
                              const float* __restrict__ Bt, const float* __restrict__ bias,
                              float* __restrict__ C, long bsC, int ldc,
                              int M, int N, int K,
                              int tilesM, int tilesN64, int batch, int act) {
  int gw = (int)(((long)blockIdx.x * blockDim.x + threadIdx.x) >> 5);
  int tilesTot = tilesM * tilesN64;
  if (gw >= tilesTot * batch) return;            // whole-wave uniform exit
  int bidx = gw / tilesTot;
  int tile = gw - bidx * tilesTot;
  int tm = tile / tilesN64;
  int tn = tile - tm * tilesN64;

  int lane = threadIdx.x & 31;
  int half = lane >> 4;                          // 0: K=k0,k0+1  1: K=k0+2,k0+3
  int l16  = lane & 15;

  int rowA = tm * 16 + l16;
  bool rowOK = rowA < M;
  long aRow = rowOK ? (long)rowA * Ars : 0;
  float mA  = rowOK ? 1.f : 0.f;

  int   col[4];
  long  cc[4];
  float mB[4];
#pragma unroll
  for (int u = 0; u < 4; ++u) {
    col[u] = tn * 64 + u * 16 + l16;
    bool ok = col[u] < N;
    cc[u] = ok ? col[u] : 0;
    mB[u] = ok ? 1.f : 0.f;
  }

  const float* Ab = A + (long)bidx * bsA;
  const v2f*   Bp = (const v2f*)Bt;
  v8f acc0 = {}, acc1 = {}, acc2 = {}, acc3 = {};

  bool interior = (tm * 16 + 16 <= M) && (tn * 64 + 64 <= N);  // wave-uniform
  if (Acs == 1) {
    if (interior) gemm_kloop<true, false>(Ab, aRow, Acs, Bp, N, K, half, mA, cc, mB, acc0, acc1, acc2, acc3);
    else          gemm_kloop<true, true >(Ab, aRow, Acs, Bp, N, K, half, mA, cc, mB, acc0, acc1, acc2, acc3);
  } else {
    if (interior) gemm_kloop<false, false>(Ab, aRow, Acs, Bp, N, K, half, mA, cc, mB, acc0, acc1, acc2, acc3);
    else          gemm_kloop<false, true >(Ab, aRow, Acs, Bp, N, K, half, mA, cc, mB, acc0, acc1, acc2, acc3);
  }

  float* Cb = C + (long)bidx * bsC;
  v8f accs[4] = {acc0, acc1, acc2, acc3};
#pragma unroll
  for (int u = 0; u < 4; ++u) {
    if (col[u] >= N) continue;
    float bb = bias ? bias[col[u]] : 0.f;
#pragma unroll
    for (int j = 0; j < 8; ++j) {
      int mrow = tm * 16 + j + (half << 3);
      if (mrow < M) {
        float v = accs[u][j] + bb;
        if      (act == 1) v = 1.f / (1.f + __expf(-v));   // sigmoid
        else if (act == 2) v = tanhf(v);                   // tanh
        else if (act == 3) v = fmaxf(v, 0.f);              // relu
        Cb[(long)mrow * ldc + col[u]] = v;
      }
    }
  }
}

// ---------------------------------------------------------------------------
__global__ void gather_kv(const int* __restrict__ q, const int* __restrict__ r,
                          const float* __restrict__ k_emb, const float* __restrict__ v_emb,
                          float* __restrict__ kb, float* __restrict__ vb) {
  long row = blockIdx.x;                 // b*n + t
  int tid = threadIdx.x;
  int qi = q[row];
  long x = (long)qi + (long)NUMQ * r[row];
  if (tid < DIMS) {
    kb[row * DIMS + tid] = k_emb[(long)qi * DIMS + tid];
    vb[row * DIMS + tid] = v_emb[x * DIMS + tid];
  }
}

// ---------------------------------------------------------------------------
__global__ void softmax50(const float* __restrict__ logits, float* __restrict__ wout,
                          int rows) {
  int row = blockIdx.x * blockDim.x + threadIdx.x;
  if (row >= rows) return;
  const float* L = logits + (long)row * SLOTS;
  float mx = -3.0e38f;
  for (int i = 0; i < SLOTS; ++i) mx = fmaxf(mx, L[i]);
  float buf[SLOTS];
  float sum = 0.f;
  for (int i = 0; i < SLOTS; ++i) { float v = __expf(L[i] - mx); buf[i] = v; sum += v; }
  float inv = 1.f / sum;
  float* W = wout + (long)row * SLOTS;
  for (int i = 0; i < SLOTS; ++i) W[i] = buf[i] * inv;
}

// ---------------------------------------------------------------------------
// Sequential memory scan: one block per batch, Mv resident in LDS (40KB).
// Mv0 (50x200 f32, 10000 contiguous dwords) is DMA'd into LDS by the Tensor
// Data Mover; wave 0 issues the D# (ISA ch.8), waits s_wait_tensorcnt 0, then
// the workgroup barrier publishes LDS to all waves.
// ---------------------------------------------------------------------------
__global__ void scan_kernel(const float* __restrict__ w,    // [B, n, m]
                            const float* __restrict__ e,    // [B, n, d]
                            const float* __restrict__ a,    // [B, n, d]
                            const float* __restrict__ Mv0,  // [m, d]
                            float* __restrict__ MvOut,      // [B, n+1, m, d]
                            float* __restrict__ readOut) {  // [B, n, d]
  __shared__ float Mv_s[SLOTS * DIMS];
  __shared__ float w_s[SLOTS];
  int b = blockIdx.x;
  int tid = threadIdx.x;

  if (tid < 32) {                                // one wave drives the TDM
    unsigned long long ga = (unsigned long long)Mv0;
    unsigned lds_off = (unsigned)(unsigned long long)&Mv_s[0];  // ADDR[31:0] = LDS offset
    v4u g0;
    g0.x = 1u;                                   // count=1 (valid), is_restore=0
    g0.y = lds_off;                              // lds_addr (bytes)
    g0.z = (unsigned)(ga & 0xFFFFFFFFull);       // global_addr[31:0]
    g0.w = (unsigned)((ga >> 32) & 0x01FFFFFFull) | (2u << 30);  // addr[56:32] | type=2
    v8i g1 = {};
    const unsigned ndw = SLOTS * DIMS;           // 10000 elements (fits 16-bit tile_dim0)
    g1[0] = (int)(2u << 16);                     // data_size=2 -> 4 bytes; mask=0; no pad
    g1[1] = (int)((ndw & 0xFFFFu) << 16);        // tensor_dim0[15:0]
    g1[2] = (int)(((ndw >> 16) & 0xFFFFu) | (1u << 16));  // tensor_dim0[31:16] | tensor_dim1=1
    g1[3] = (int)((ndw & 0xFFFFu) << 16);        // tile_dim0 = 10000
    g1[4] = 0;                                   // tile_dim1=0 (unused), tile_dim2=0
    g1[5] = (int)ndw;                            // tensor_dim0_stride[31:0]
    g1[6] = 0;
    g1[7] = 0;
    v4i g2 = {};
    v4i g3 = {};
    v8i g4 = {};                                 // 6-arg toolchain: extra group, unused
    __builtin_amdgcn_tensor_load_to_lds(g0, g1, g2, g3, g4, 0);
    __builtin_amdgcn_s_wait_tensorcnt(0);
  }
  __syncthreads();

  for (int t = 0; t < DIMS; ++t) {               // n == DIMS sequential steps
    if (tid < SLOTS) w_s[tid] = w[((long)b * DIMS + t) * SLOTS + tid];
    __syncthreads();
    for (int j = tid; j < DIMS; j += blockDim.x) {
      long rowOff = ((long)b * DIMS + t) * DIMS + j;
      float ej = e[rowOff];
      float aj = a[rowOff];
      float sum = 0.f;
      long outBase = (((long)b * (DIMS + 1) + t) * SLOTS) * DIMS + j;
#pragma unroll 5
      for (int s = 0; s < SLOTS; ++s) {
        float mv = Mv_s[s * DIMS + j];
        MvOut[outBase + (long)s * DIMS] = mv;    // pre-update state, coalesced in j
        float ws = w_s[s];
        sum += ws * mv;
        Mv_s[s * DIMS + j] = mv * (1.f - ws * ej) + ws * aj;
      }
      readOut[rowOff] = sum;
    }
    __syncthreads();
  }
  for (int i = tid; i < SLOTS * DIMS; i += blockDim.x)   // final state at index n
    MvOut[(((long)b * (DIMS + 1) + DIMS) * SLOTS) * DIMS + i] = Mv_s[i];
}

// ---------------------------------------------------------------------------
__global__ void concat2(const float* __restrict__ X, const float* __restrict__ Y,
                        float* __restrict__ Z) {
  long row = blockIdx.x;
  int tid = threadIdx.x;
  if (tid < DIMS) {
    Z[row * 2 * DIMS + tid]        = X[row * DIMS + tid];
    Z[row * 2 * DIMS + DIMS + tid] = Y[row * DIMS + tid];
  }
}

// ---------------------------------------------------------------------------
// p[b,t] = sigmoid( f[b,t,:].Wp[0:200] + f_l[b,:,t].Wp[200:400] + bp ), wave/row
// ---------------------------------------------------------------------------
__global__ void final_p(const float* __restrict__ f, const float* __restrict__ fl,
                        const float* __restrict__ Wp, const float* __restrict__ bp,
                        float* __restrict__ p) {
  int wave = (int)(((long)blockIdx.x * blockDim.x + threadIdx.x) >> 5);
  int lane = threadIdx.x & 31;
  if (wave >= NB * DIMS) return;
  int b = wave / DIMS, t = wave - b * DIMS;
  float partial = 0.f;
  for (int j = lane; j < DIMS; j += 32) {
    partial += f[((long)b * DIMS + t) * DIMS + j] * Wp[j];
    partial += fl[((long)b * DIMS + j) * DIMS + t] * Wp[DIMS + j];   // transposed f_l
  }
  for (int off = 16; off > 0; off >>= 1) partial += __shfl_xor(partial, off, 32);
  if (lane == 0) p[(long)b * DIMS + t] = 1.f / (1.f + __expf(-(partial + bp[0])));
}

// ---------------------------------------------------------------------------
static inline void launch_gemm(hipStream_t s,
                               const float* A, long bsA, int Ars, int Acs,
                               const float* Bt, const float* bias,
                               float* C, long bsC, int ldc,
                               int M, int N, int K, int batch, int act) {
  int tilesM = (M + 15) / 16, tilesN64 = (N + 63) / 64;
  long waves = (long)tilesM * tilesN64 * batch;
  int block = 256;                                  // 8 wave32 per block
  long blocks = (waves * 32 + block - 1) / block;
  gemm_wmma_f32<<<dim3((unsigned)blocks), dim3(block), 0, s>>>(
      A, bsA, Ars, Acs, Bt, bias, C, bsC, ldc, M, N, K, tilesM, tilesN64, batch, act);
}

static inline void launch_transpose(hipStream_t s, const float* W, float* Wt,
                                    int N, int K) {
  long elems = (long)N * K;
  transpose_w<<<dim3((unsigned)((elems + 255) / 256)), dim3(256), 0, s>>>(W, Wt, N, K);
}

extern "C" void kernel_launch(void* const* d_in, const int* in_sizes, int n_in,
                              void* d_out, int out_size, void* d_ws, size_t ws_size,
                              hipStream_t stream) {
  const int*   q      = (const int*)  d_in[0];
  const int*   r      = (const int*)  d_in[1];
  const float* bert   = (const float*)d_in[2];   // [64,512,768]
  const float* k_emb  = (const float*)d_in[3];
  const float* v_emb  = (const float*)d_in[4];
  const float* Mk     = (const float*)d_in[5];
  const float* Mv0    = (const float*)d_in[6];
  const float* W_at   = (const float*)d_in[7];
  const float* b_at   = (const float*)d_in[8];
  const float* W_at2  = (const float*)d_in[9];
  const float* b_at2  = (const float*)d_in[10];
  const float* W_fus  = (const float*)d_in[11];
  const float* b_fus  = (const float*)d_in[12];
  const float* W_e    = (const float*)d_in[13];
  const float* b_e    = (const float*)d_in[14];
  const float* W_a    = (const float*)d_in[15];
  const float* b_a    = (const float*)d_in[16];
  const float* W_f    = (const float*)d_in[17];
  const float* b_f    = (const float*)d_in[18];
  const float* W_p    = (const float*)d_in[19];
  const float* b_p    = (const float*)d_in[20];

  // outputs: p [64*200] | Mv [64*201*50*200] | w [64*200*50]
  float* p_out  = (float*)d_out;
  float* Mv_out = p_out + 12800;
  float* w_out  = Mv_out + 128640000L;

  // workspace layout (floats), lifetime-based reuse (all offsets even -> 8B align)
  float* ws     = (float*)d_ws;
  float* tW_at  = ws;                    // 768*200 = 153,600
  float* tW_at2 = tW_at  + 153600;       // 512*200 = 102,400
  float* tMk    = tW_at2 + 102400;       // 200*50  =  10,000
  float* tW_e   = tMk    + 10000;        // 200*200 =  40,000
  float* tW_a   = tW_e   + 40000;        //            40,000
  float* tW_f   = tW_a   + 40000;        // 400*200 =  80,000
  float* tW_fus = tW_f   + 80000;        //            80,000
  float* em     = tW_fus + 80000;        // 32768*200 = 6,553,600
  float* emat   = em     + 6553600;      // 2,560,000
  float* kbuf   = emat   + 2560000;      // 2,560,000
  float* vbuf   = kbuf   + 2560000;      // 2,560,000
  float* wlog   = vbuf   + 2560000;      //   640,000
  float* ebuf   = wlog   + 640000;       // 2,560,000
  float* abuf   = ebuf   + 2560000;      // 2,560,000
  float* rbuf   = abuf   + 2560000;      // 2,560,000
  float* vabuf  = rbuf   + 2560000;      // 5,120,000   (total ~113 MB)
  float* rkbuf  = em;                    // reuse: em dead after em_at GEMM
  float* fbuf   = ebuf;                  // reuse: e dead after scan
  float* flbuf  = abuf;                  // reuse: a dead after scan

  // 0) transpose + k-pair-pack all weights for b64 fragment loads
  launch_transpose(stream, W_at,  tW_at,  200, 768);
  launch_transpose(stream, W_at2, tW_at2, 200, 512);
  launch_transpose(stream, Mk,    tMk,     50, 200);
  launch_transpose(stream, W_e,   tW_e,   200, 200);
  launch_transpose(stream, W_a,   tW_a,   200, 200);
  launch_transpose(stream, W_f,   tW_f,   200, 400);
  launch_transpose(stream, W_fus, tW_fus, 200, 400);

  // 1) em = bert @ W_at^T + b_at           [32768,200], K=768
  launch_gemm(stream, bert, 0, 768, 1, tW_at, b_at, em, 0, 200,
              32768, 200, 768, 1, 0);
  // 2) em_at[b] = em[b]^T @ W_at2^T + b_at2  batched [200,200], K=512 (A transposed via strides)
  launch_gemm(stream, em, 512L * 200, 1, 200, tW_at2, b_at2, emat, 40000, 200,
              200, 200, 512, NB, 0);
  // 3) gathers
  gather_kv<<<dim3(12800), dim3(256), 0, stream>>>(q, r, k_emb, v_emb, kbuf, vbuf);
  // 4) w logits = k @ Mk^T                 [12800,50], K=200
  launch_gemm(stream, kbuf, 0, 200, 1, tMk, nullptr, wlog, 0, 50,
              12800, 50, 200, 1, 0);
  // 5) softmax -> w output
  softmax50<<<dim3((12800 + 255) / 256), dim3(256), 0, stream>>>(wlog, w_out, 12800);
  // 6) e = sigmoid(v @ W_e^T + b_e)
  launch_gemm(stream, vbuf, 0, 200, 1, tW_e, b_e, ebuf, 0, 200,
              12800, 200, 200, 1, 1);
  // 7) a = tanh(v @ W_a^T + b_a)
  launch_gemm(stream, vbuf, 0, 200, 1, tW_a, b_a, abuf, 0, 200,
              12800, 200, 200, 1, 2);
  // 8) sequential scan (TDM Mv0 preload, Mv stream + read), one block per batch
  scan_kernel<<<dim3(NB), dim3(256), 0, stream>>>(w_out, ebuf, abuf, Mv0, Mv_out, rbuf);
  // 9) concats
  concat2<<<dim3(12800), dim3(256), 0, stream>>>(rbuf, kbuf, rkbuf);
  concat2<<<dim3(12800), dim3(256), 0, stream>>>(vbuf, emat, vabuf);
  // 10) f = tanh([read,k] @ W_f^T + b_f)   K=400
  launch_gemm(stream, rkbuf, 0, 400, 1, tW_f, b_f, fbuf, 0, 200,
              12800, 200, 400, 1, 2);
  // 11) f_l = relu([v,em_at] @ W_fus^T + b_fus)
  launch_gemm(stream, vabuf, 0, 400, 1, tW_fus, b_fus, flbuf, 0, 200,
              12800, 200, 400, 1, 3);
  // 12) p
  final_p<<<dim3((12800 * 32 + 255) / 256), dim3(256), 0, stream>>>(
      fbuf, flbuf, W_p, b_p, p_out);
}